// TransitionCRF_2250562863910
// MI455X (gfx1250) — compile-verified
//
#include <hip/hip_runtime.h>
#include <stdint.h>

#define T_LEN 1024
#define P_DIM 32
#define HOME_IDX 0
#define PF 16   // prefetch distance (time steps)

// One wave32 per batch element. lane == purpose state p.
// LDS layout: psi[(T-1)*64] bytes | pathY[T] bytes | Astage[32*32] floats
__launch_bounds__(32)
__global__ void viterbi32_kernel(const float* __restrict__ U,
                                 const float* __restrict__ A,
                                 const float* __restrict__ bias,
                                 int* __restrict__ out) {
  extern __shared__ unsigned char smem[];
  unsigned char* psi   = smem;                                   // (T-1)*64 B
  unsigned char* pathY = smem + (T_LEN - 1) * 64;                // T bytes
  float* Ast = (float*)(smem + (T_LEN - 1) * 64 + T_LEN);        // 1024 floats

  const int lane = threadIdx.x;   // 0..31
  const int b    = blockIdx.x;

  // ---- Stage A (32x32 f32, row-major) into LDS via async tensor path ----
  {
    uint64_t abase = (uint64_t)(const void*)A;
    unsigned ldsA  = (unsigned)(size_t)(void*)Ast;  // low 32 bits = LDS offset
    #pragma unroll
    for (int i = 0; i < 8; ++i) {
      unsigned off = (unsigned)(i * 2048 + lane * 16);  // 16B per lane per iter
      unsigned dst = ldsA + off;
      asm volatile("global_load_async_to_lds_b128 %0, %1, %2"
                   :: "v"(dst), "v"(off), "s"(abase) : "memory");
    }
    asm volatile("s_wait_asynccnt 0" ::: "memory");
  }
  __syncthreads();

  // Each lane keeps its column: acol[q] = A[q][lane]  (conflict-free LDS reads)
  float acol[32];
  #pragma unroll
  for (int q = 0; q < 32; ++q) acol[q] = Ast[q * 32 + lane];

  const float NEGV  = -3.4028234663852886e38f * 0.25f;  // finfo(f32).min / 4
  const float biasv = bias[lane];
  const float* Ub   = U + ((size_t)b * T_LEN * P_DIM) + lane;

  // t = 0 : v=0 only at home, v=1 only at non-home
  float u0 = Ub[0] + biasv;
  float d0 = (lane == HOME_IDX) ? u0 : NEGV;
  float d1 = (lane == HOME_IDX) ? NEGV : u0;

  for (int t = 1; t < T_LEN; ++t) {
    if (t + PF < T_LEN)
      __builtin_prefetch(Ub + (size_t)(t + PF) * P_DIM, 0, 3);  // global_prefetch_b8
    float ut = Ub[(size_t)t * P_DIM] + biasv;

    // max-plus over q with first-max argmax, both v-channels
    float v0 = -__builtin_inff(); int a0 = 0;
    float v1 = -__builtin_inff(); int a1 = 0;
    #pragma unroll
    for (int q = 0; q < 32; ++q) {
      float dq0 = __uint_as_float(__builtin_amdgcn_readlane(__float_as_uint(d0), q));
      float dq1 = __uint_as_float(__builtin_amdgcn_readlane(__float_as_uint(d1), q));
      float p0 = dq0 + acol[q];
      float p1 = dq1 + acol[q];
      if (p0 > v0) { v0 = p0; a0 = q; }   // strict > : first max wins (jnp.argmax)
      if (p1 > v1) { v1 = p1; a1 = q; }
    }

    bool  use1 = (v1 > v0);               // strict, matches reference
    float bval = use1 ? v1 : v0;
    int   bq   = use1 ? a1 : a0;
    int   bvf  = use1 ? 1 : 0;
    bool  home = (lane == HOME_IDX);

    float dn0 = home ? (v0 + ut) : NEGV;
    float dn1 = (home ? v1 : bval) + ut;
    int   p1i = home ? a1 : bq;
    int   vv1 = home ? 1 : bvf;

    // pack backpointers: bits[4:0]=prev purpose, bit5=prev flag
    psi[(t - 1) * 64 + lane]      = (unsigned char)a0;            // v=0 (flag 0)
    psi[(t - 1) * 64 + 32 + lane] = (unsigned char)(p1i | (vv1 << 5));

    d0 = dn0; d1 = dn1;
  }

  // last_p = argmax_p delta_T[p][1] (first max) — computed uniformly by all lanes
  float bvv = -__builtin_inff(); int bp = 0;
  #pragma unroll
  for (int q = 0; q < 32; ++q) {
    float x = __uint_as_float(__builtin_amdgcn_readlane(__float_as_uint(d1), q));
    if (x > bvv) { bvv = x; bp = q; }
  }

  __syncthreads();
  if (lane == 0) pathY[T_LEN - 1] = (unsigned char)bp;

  // Backtrack: all lanes walk redundantly (broadcast LDS reads), lane0 records.
  int y = bp, v = 1;
  for (int t = T_LEN - 2; t >= 0; --t) {
    unsigned char c = psi[t * 64 + v * 32 + y];
    y = c & 31;
    v = (c >> 5) & 1;
    if (lane == 0) pathY[t] = (unsigned char)y;
  }
  __syncthreads();

  // Coalesced int32 writeout
  int* ob = out + (size_t)b * T_LEN;
  #pragma unroll
  for (int i = 0; i < T_LEN / 32; ++i)
    ob[i * 32 + lane] = (int)pathY[i * 32 + lane];
}

extern "C" void kernel_launch(void* const* d_in, const int* in_sizes, int n_in,
                              void* d_out, int out_size, void* d_ws, size_t ws_size,
                              hipStream_t stream) {
  const float* U    = (const float*)d_in[0];
  const float* A    = (const float*)d_in[1];
  const float* bias = (const float*)d_in[2];
  int* out = (int*)d_out;

  int B = in_sizes[0] / (T_LEN * P_DIM);   // 256
  size_t shmem = (size_t)(T_LEN - 1) * 64 + T_LEN + P_DIM * P_DIM * sizeof(float);
  viterbi32_kernel<<<B, 32, shmem, stream>>>(U, A, bias, out);
}